// _GraphConv_16423954940709
// MI455X (gfx1250) — compile-verified
//
#include <hip/hip_runtime.h>
#include <hip/hip_bf16.h>

typedef float v2f __attribute__((ext_vector_type(2)));
typedef float v4f __attribute__((ext_vector_type(4)));
typedef float v8f __attribute__((ext_vector_type(8)));

#define BATCH   8192
#define JJ      17
#define FDIM    128
#define TB      16                 // batches per workgroup
#define ROWS    (TB * JJ)          // 272 rows per WG
#define MTILES  (ROWS / 16)        // 17 exact 16-row tiles
#define BN_EPS  1e-5f
#define XPITCH  132                // padded x-tile row pitch (bank-conflict-free A frags)

// ---- dynamic LDS layout (float offsets) ----
#define L_OUT   0                          // diag*M*h0 + bias   [272][128]
#define L_H1    (ROWS * FDIM)              // M*h1               [272][128]
#define L_OFFP  (2 * ROWS * FDIM)          // zero-padded OFF    [32][20]
#define L_DIAG  (L_OFFP + 32 * 20)         // diag(a)            [17]
#define L_BIAS  (L_DIAG + JJ)              // bias               [128]
#define L_XBUF  (((L_BIAS + FDIM) + 3) & ~3)  // 16B-aligned; x tiles [2][16][132]
#define LDS_FLOATS (L_XBUF + 2 * 16 * XPITCH) // 74644 floats = 298576 B < 320KB

// ---------------------------------------------------------------------------
// Kernel 0: zero the BN accumulators in workspace (ws is poisoned by harness)
// ---------------------------------------------------------------------------
__global__ void zero_ws(float* __restrict__ ws) {
    ws[threadIdx.x] = 0.0f;   // 256 floats: [0,128)=sum, [128,256)=sumsq
}

// ---------------------------------------------------------------------------
// Kernel 1: fused  h0=x@W0, h1=x@W1 (FP32 WMMA), adjacency mix as WMMA
//           (OFF zero-padded to 32x20), bias, and BN sum/sumsq accumulation.
// Grid: BATCH/TB = 512 blocks, 256 threads (8 wave32), ~292KB dynamic LDS.
// ---------------------------------------------------------------------------
__global__ __launch_bounds__(256)
void gconv_wmma(const float* __restrict__ x,     // [B,J,128]
                const float* __restrict__ W,     // [2,128,128]
                const float* __restrict__ Mmod,  // [J,128]
                const float* __restrict__ adj,   // [J,J]
                const float* __restrict__ adj2,  // [J,J]
                const float* __restrict__ bias,  // [128]
                float* __restrict__ outPre,      // [B,J,128] (pre-BN, to d_out)
                float* __restrict__ wsAccum)     // [256] sum/sumsq
{
    extern __shared__ float lds[];
    const int tid   = threadIdx.x;
    const int lane  = tid & 31;
    const int wave  = tid >> 5;                 // 0..7 -> N-tile index
    const int n0    = wave * 16 + (lane & 15);  // this lane's output column
    const int khalf = lane >> 4;                // 0 or 1 (K/row half of frag)
    const int arow  = lane & 15;                // A-fragment row within tile
    const int rowBaseWG = blockIdx.x * ROWS;    // global flattened (b*17+j) base

    // -----------------------------------------------------------------------
    // Async stage of one 16x128 x tile into LDS (512 x 16B chunks, 2/thread).
    // GVS mode: mem = SADDR64(x) + VADDR32(byte offset); LDS addr per lane.
    // -----------------------------------------------------------------------
    auto stage = [&](int t) {
        const int bufo = L_XBUF + (t & 1) * (16 * XPITCH);
#pragma unroll
        for (int p = 0; p < 2; ++p) {
            const int ch  = tid + p * 256;          // 0..511
            const int row = ch >> 5;                // 0..15
            const int c4  = (ch & 31) << 2;         // float col, x4 -> 16B chunks
            const unsigned laddr = (unsigned)((bufo + row * XPITCH + c4) * 4);
            const unsigned goff  =
                (unsigned)(((rowBaseWG + t * 16 + row) * FDIM + c4) * 4);
            asm volatile("global_load_async_to_lds_b128 %0, %1, %2"
                         :: "v"(laddr), "v"(goff), "s"(x) : "memory");
        }
    };

    stage(0);   // kick off tile 0 before table init (overlap)

    // ---- build zero-padded symmetrized adjacency tables in LDS ----
    for (int t = tid; t < 32 * 20; t += 256) {
        const int i = t / 20, j = t - (t / 20) * 20;
        float v = 0.0f;
        if (i < JJ && j < JJ && i != j)
            v = 0.5f * (adj[i * JJ + j] + adj2[i * JJ + j] +
                        adj[j * JJ + i] + adj2[j * JJ + i]);
        lds[L_OFFP + t] = v;
    }
    if (tid < JJ)   lds[L_DIAG + tid] = adj[tid * JJ + tid] + adj2[tid * JJ + tid];
    if (tid < FDIM) lds[L_BIAS + tid] = bias[tid];

    // ---- preload B fragments for W0/W1 (K=128 -> 32 steps, 128 VGPRs) ----
    const float* W0 = W;
    const float* W1 = W + FDIM * FDIM;
    v2f bw0[32], bw1[32];
#pragma unroll
    for (int kk = 0; kk < 32; ++kk) {
        const int kb = kk * 4 + khalf * 2;
        bw0[kk].x = W0[kb * FDIM + n0];
        bw0[kk].y = W0[(kb + 1) * FDIM + n0];
        bw1[kk].x = W1[kb * FDIM + n0];
        bw1[kk].y = W1[(kb + 1) * FDIM + n0];
    }

    // =======================================================================
    // Phase 1: double-buffered async x staging + FP32 WMMA GEMMs.
    // =======================================================================
    for (int mt = 0; mt < MTILES; ++mt) {
        if (mt + 1 < MTILES) {
            stage(mt + 1);
            asm volatile("s_wait_asynccnt 0x2" ::: "memory"); // tile mt landed
        } else {
            asm volatile("s_wait_asynccnt 0x0" ::: "memory");
        }
        __syncthreads();   // staged tile (all waves) + tables visible

        const float* xb = &lds[L_XBUF + (mt & 1) * (16 * XPITCH)];
        v8f acc0 = {};
        v8f acc1 = {};
#pragma unroll
        for (int kk = 0; kk < 32; ++kk) {
            const int kb = kk * 4 + khalf * 2;
            const v2f a = *(const v2f*)(xb + arow * XPITCH + kb); // ds_load_b64
            acc0 = __builtin_amdgcn_wmma_f32_16x16x4_f32(
                       false, a, false, bw0[kk], (short)0, acc0, false, false);
            acc1 = __builtin_amdgcn_wmma_f32_16x16x4_f32(
                       false, a, false, bw1[kk], (short)0, acc1, false, false);
        }

        // Scatter C tiles to LDS, folding modulation M, diag scaling and bias.
#pragma unroll
        for (int r = 0; r < 8; ++r) {
            const int grow = mt * 16 + r + khalf * 8;     // 0..271 within WG
            const int j    = grow % JJ;                   // joint index
            const float m  = Mmod[j * FDIM + n0];
            lds[L_OUT + grow * FDIM + n0] =
                lds[L_DIAG + j] * (acc0[r] * m) + lds[L_BIAS + n0];
            lds[L_H1 + grow * FDIM + n0] = acc1[r] * m;
        }
        __syncthreads();   // protect xbuf[(mt)&1] before it is re-staged
    }

    // =======================================================================
    // Phase 2: adjacency mix as WMMA:  OUT[b](17x128) += OFFp(32x20) @ H1[b].
    // K padded to 20 with zero OFF columns -> OOB H1/C reads are neutralized;
    // invalid output rows (i>=17) are computed but never scattered.
    // =======================================================================
    float lsum = 0.0f, lsq = 0.0f;
    for (int bb = 0; bb < TB; ++bb) {
#pragma unroll
        for (int mi = 0; mi < 2; ++mi) {
            v8f c;
#pragma unroll
            for (int r = 0; r < 8; ++r) {
                const int lrow = bb * JJ + mi * 16 + r + khalf * 8;
                c[r] = lds[L_OUT + lrow * FDIM + n0];   // finite junk if OOB row
            }
#pragma unroll
            for (int ks = 0; ks < 5; ++ks) {
                const int kb = ks * 4 + khalf * 2;      // K rows kb, kb+1
                v2f a, b;
                a.x = lds[L_OFFP + (mi * 16 + arow) * 20 + kb];
                a.y = lds[L_OFFP + (mi * 16 + arow) * 20 + kb + 1];
                b.x = lds[L_H1 + (bb * JJ + kb) * FDIM + n0];
                b.y = lds[L_H1 + (bb * JJ + kb + 1) * FDIM + n0];
                c = __builtin_amdgcn_wmma_f32_16x16x4_f32(
                        false, a, false, b, (short)0, c, false, false);
            }
#pragma unroll
            for (int r = 0; r < 8; ++r) {
                const int i = mi * 16 + r + khalf * 8;
                if (i < JJ) {
                    const float v = c[r];
                    outPre[(size_t)(rowBaseWG + bb * JJ + i) * FDIM + n0] = v;
                    lsum += v;
                    lsq  = fmaf(v, v, lsq);
                }
            }
        }
    }
    atomicAdd(&wsAccum[n0], lsum);
    atomicAdd(&wsAccum[FDIM + n0], lsq);
}

// ---------------------------------------------------------------------------
// Kernel 2: turn sum/sumsq into per-channel scale/shift
// ---------------------------------------------------------------------------
__global__ void bn_finalize(float* __restrict__ ws,
                            const float* __restrict__ gamma,
                            const float* __restrict__ beta) {
    const int c = threadIdx.x;   // 128 threads
    const float invN = 1.0f / (float)(BATCH * JJ);
    const float mean = ws[c] * invN;
    const float var  = ws[FDIM + c] * invN - mean * mean;
    const float scale = gamma[c] * rsqrtf(var + BN_EPS);
    ws[2 * FDIM + c] = scale;
    ws[3 * FDIM + c] = beta[c] - mean * scale;
}

// ---------------------------------------------------------------------------
// Kernel 3: in-place normalize + ReLU, float4 vectorized (HBM streaming pass)
// ---------------------------------------------------------------------------
__global__ __launch_bounds__(256)
void bn_relu(float* __restrict__ out, const float* __restrict__ ws) {
    const int idx = (blockIdx.x * 256 + threadIdx.x) * 4;
    const int c   = idx & 127;                 // 4-aligned, no channel wrap
    v4f v  = *(v4f*)(out + idx);
    const v4f sc = *(const v4f*)(ws + 2 * FDIM + c);
    const v4f sh = *(const v4f*)(ws + 3 * FDIM + c);
    v.x = fmaxf(fmaf(v.x, sc.x, sh.x), 0.0f);
    v.y = fmaxf(fmaf(v.y, sc.y, sh.y), 0.0f);
    v.z = fmaxf(fmaf(v.z, sc.z, sh.z), 0.0f);
    v.w = fmaxf(fmaf(v.w, sc.w, sh.w), 0.0f);
    *(v4f*)(out + idx) = v;
}

// ---------------------------------------------------------------------------
extern "C" void kernel_launch(void* const* d_in, const int* in_sizes, int n_in,
                              void* d_out, int out_size, void* d_ws, size_t ws_size,
                              hipStream_t stream) {
    (void)in_sizes; (void)n_in; (void)out_size; (void)ws_size;
    const float* x     = (const float*)d_in[0];
    const float* W     = (const float*)d_in[1];
    const float* Mmod  = (const float*)d_in[2];
    const float* adj   = (const float*)d_in[3];
    const float* adj2  = (const float*)d_in[4];
    const float* bias  = (const float*)d_in[5];
    const float* gamma = (const float*)d_in[6];
    const float* beta  = (const float*)d_in[7];
    float* out = (float*)d_out;
    float* ws  = (float*)d_ws;

    const size_t ldsBytes = (size_t)LDS_FLOATS * sizeof(float);   // ~292KB
    (void)hipFuncSetAttribute((const void*)gconv_wmma,
                              hipFuncAttributeMaxDynamicSharedMemorySize,
                              (int)ldsBytes);

    zero_ws<<<1, 256, 0, stream>>>(ws);
    gconv_wmma<<<BATCH / TB, 256, ldsBytes, stream>>>(x, W, Mmod, adj, adj2,
                                                      bias, out, ws);
    bn_finalize<<<1, 128, 0, stream>>>(ws, gamma, beta);
    bn_relu<<<(BATCH * JJ * FDIM) / (256 * 4), 256, 0, stream>>>(out, ws);
}